// GINNet_48541720379897
// MI455X (gfx1250) — compile-verified
//
#include <hip/hip_runtime.h>

// GIN 2-layer network, MI455X (gfx1250), wave32.
//   x    = emb[x_indices]                        [N,128]
//   agg1 = x + scatter_sum(x[src] -> dst)        (global_atomic_add_f32, L2-resident)
//   x1   = relu( relu(agg1@W1a+b1a) @ W1b + b1b) (fused WMMA f32 kernel)
//   agg2 = x1 + scatter_sum(x1[src] -> dst)
//   out  =       relu(agg2@W2a+b2a) @ W2b + b2b  [N,64]
//
// GEMMs: V_WMMA_F32_16X16X4_F32 (exact fp32; ~6 GFLOP total, far below the
// WMMA ceiling -- workload is L2/atomic bound). Weights are DMA'd into LDS
// once per block with TENSOR_LOAD_TO_LDS (TDM) and drained via TENSORcnt.

typedef __attribute__((ext_vector_type(2))) float v2f;
typedef __attribute__((ext_vector_type(8))) float v8f;
typedef __attribute__((ext_vector_type(4))) unsigned int v4u;
typedef __attribute__((ext_vector_type(4))) int v4i;
typedef __attribute__((ext_vector_type(8))) int v8i;

#define DIM 128

// ---------------------------------------------------------------------------
// Generic LDS pointer -> LDS byte offset (ISA 10.2: LDS aperture address
// truncates to the LDS offset in its low 32 bits).
__device__ static inline unsigned int lds_byte_off(const void* p) {
  return (unsigned int)(unsigned long long)(uintptr_t)p;
}

// TDM: async 2-D tile load (rows x cols f32, row stride in elements) from
// global memory into LDS. Descriptor per CDNA5 ISA ch.8 (D# groups 0/1;
// groups 2/3 zero => <=2-D tensor). Tracked by TENSORcnt.
__device__ static inline void tdm_load_2d_f32(const void* gptr, unsigned int lds_off,
                                              unsigned int rows, unsigned int cols,
                                              unsigned int row_stride) {
  unsigned long long ga = (unsigned long long)(uintptr_t)gptr;
  v4u g0;
  g0[0] = 1u;                                        // count=1, user descriptor
  g0[1] = lds_off;                                   // lds_addr (bytes)
  g0[2] = (unsigned int)(ga & 0xFFFFFFFFu);          // global_addr[31:0]
  g0[3] = (unsigned int)((ga >> 32) & 0x01FFFFFFu)   // global_addr[56:32]
        | (2u << 30);                                // type=2 ("image")
  v8i g1;
  g1[0] = (int)(2u << 16);                           // wg_mask=0, data_size=2 (4B)
  g1[1] = (int)((cols & 0xFFFFu) << 16);             // tensor_dim0[15:0]
  g1[2] = (int)(((cols >> 16) & 0xFFFFu)             // tensor_dim0[31:16]
        | ((rows & 0xFFFFu) << 16));                 // tensor_dim1[15:0]
  g1[3] = (int)(((rows >> 16) & 0xFFFFu)             // tensor_dim1[31:16]
        | ((cols & 0xFFFFu) << 16));                 // tile_dim0 = cols
  g1[4] = (int)(rows & 0xFFFFu);                     // tile_dim1 = rows, tile_dim2=0
  g1[5] = (int)row_stride;                           // tensor_dim0_stride[31:0]
  g1[6] = 0;                                         // dim0_stride hi, dim1_stride lo
  g1[7] = 0;                                         // dim1_stride hi
  v4i z4 = {0, 0, 0, 0};
#if __clang_major__ >= 23
  v8i z8 = {0, 0, 0, 0, 0, 0, 0, 0};
  __builtin_amdgcn_tensor_load_to_lds(g0, g1, z4, z4, z8, 0);
#else
  __builtin_amdgcn_tensor_load_to_lds(g0, g1, z4, z4, 0);
#endif
}

// ---------------------------------------------------------------------------
// Init: x = emb[x_indices], agg = x. 32 lanes/row, float4 per lane.
__global__ __launch_bounds__(256) void gather_init(
    const float* __restrict__ emb, const int* __restrict__ xi,
    float* __restrict__ x, float* __restrict__ agg, int n) {
  int t = blockIdx.x * blockDim.x + threadIdx.x;
  int row = t >> 5;
  int c = (t & 31) * 4;
  if (row >= n) return;
  const float4 v = *(const float4*)(emb + (size_t)xi[row] * DIM + c);
  *(float4*)(x   + (size_t)row * DIM + c) = v;
  *(float4*)(agg + (size_t)row * DIM + c) = v;
}

__global__ __launch_bounds__(256) void copy_f4(
    const float4* __restrict__ src, float4* __restrict__ dst, int n4) {
  int i = blockIdx.x * blockDim.x + threadIdx.x;
  if (i < n4) dst[i] = src[i];
}

// ---------------------------------------------------------------------------
// Scatter-sum: agg[dst] += x[src] per edge. 32 lanes/edge, float4/lane.
// unsafeAtomicAdd -> global_atomic_add_f32 (no CAS loop).
__global__ __launch_bounds__(256) void scatter_add(
    const float* __restrict__ x, const int* __restrict__ ei,
    float* __restrict__ agg, int E) {
  int t = blockIdx.x * blockDim.x + threadIdx.x;
  int e = t >> 5;
  if (e >= E) return;
  int lane = t & 31;
  int src = ei[e];      // ei[0][e]
  int dst = ei[E + e];  // ei[1][e]
  const float4 v = *(const float4*)(x + (size_t)src * DIM + lane * 4);
  float* p = agg + (size_t)dst * DIM + lane * 4;
  unsafeAtomicAdd(p + 0, v.x);
  unsafeAtomicAdd(p + 1, v.y);
  unsafeAtomicAdd(p + 2, v.z);
  unsafeAtomicAdd(p + 3, v.w);
}

// ---------------------------------------------------------------------------
// Fused GIN MLP: out = [relu]( relu(agg @ Wa + ba) @ Wb + bb )
// 256 threads = 8 waves; each wave owns a 16-row tile (block covers 128 rows).
// Weights staged global->LDS via TDM (Wa, then Wb reusing the same buffer);
// GEMM1 result staged through LDS (sH) to re-lay it out as GEMM2's A operand.
//
// WMMA f32 16x16x4 fragment layouts (ISA 7.12.2):
//   A 16x4:  lanes 0-15 -> K{0,1}; lanes 16-31 -> K{2,3}; row = lane%16
//   B 4x16:  lanes 0-15 -> rows K0,K1; lanes 16-31 -> rows K2,K3; col = lane%16
//   C/D:     VGPR i: lanes 0-15 -> M=i, lanes 16-31 -> M=8+i; N = lane%16
template <int COUT, bool RELU_OUT>
__global__ __launch_bounds__(256) void gin_mlp(
    const float* __restrict__ agg,  // [n,128]
    const float* __restrict__ Wa,   // [128,128]
    const float* __restrict__ ba,   // [128]
    const float* __restrict__ Wb,   // [128,COUT]
    const float* __restrict__ bb,   // [COUT]
    float* __restrict__ out,        // [n,COUT]
    int n) {
  __shared__ float sH[128 * DIM];  // 64 KB: GEMM1 output staging
  __shared__ float sW[128 * DIM];  // 64 KB: weight tile (Wa, then Wb)

  const int lane = threadIdx.x & 31;
  const int wave = threadIdx.x >> 5;  // 0..7
  const int rowBase = blockIdx.x * 128 + wave * 16;
  const int half = lane >> 4;  // 0 -> K pair (0,1), 1 -> K pair (2,3)
  const int l16 = lane & 15;

  // Stage Wa into LDS with one TDM DMA (wave 0 issues + drains TENSORcnt).
  if (wave == 0) {
    tdm_load_2d_f32(Wa, lds_byte_off(sW), 128, DIM, DIM);
    __builtin_amdgcn_s_wait_tensorcnt(0);
  }
  __syncthreads();

  // ---------------- GEMM1: h = relu(agg @ Wa + ba) ----------------
  v8f acc[DIM / 16];
#pragma unroll
  for (int nt = 0; nt < DIM / 16; ++nt) acc[nt] = (v8f)0.f;

  // Clamp OOB rows instead of branching in the k-loop (their outputs are
  // never stored, so reading a duplicated valid row is harmless).
  const int arow = rowBase + l16;
  const float* __restrict__ arowp =
      agg + (size_t)(arow < n ? arow : n - 1) * DIM;

  for (int kt = 0; kt < DIM / 4; ++kt) {
    const int k0 = kt * 4 + half * 2;
    const float2 t = *(const float2*)(arowp + k0);
    v2f a;
    a.x = t.x; a.y = t.y;
#pragma unroll
    for (int nt = 0; nt < DIM / 16; ++nt) {
      v2f b;
      b.x = sW[k0 * DIM + nt * 16 + l16];
      b.y = sW[(k0 + 1) * DIM + nt * 16 + l16];
      acc[nt] = __builtin_amdgcn_wmma_f32_16x16x4_f32(
          false, a, false, b, (short)0, acc[nt], false, false);
    }
  }

  // bias + relu, stage h into LDS (row-major 16x128 tile per wave)
#pragma unroll
  for (int nt = 0; nt < DIM / 16; ++nt) {
    const int col = nt * 16 + l16;
    const float bias = ba[col];
#pragma unroll
    for (int i = 0; i < 8; ++i) {
      float v = acc[nt][i] + bias;
      v = fmaxf(v, 0.f);
      sH[(wave * 16 + half * 8 + i) * DIM + col] = v;
    }
  }
  __syncthreads();  // sH published; all Wa reads retired -> sW reusable

  // Stage Wb into LDS (reuse sW).
  if (wave == 0) {
    tdm_load_2d_f32(Wb, lds_byte_off(sW), 128, COUT, COUT);
    __builtin_amdgcn_s_wait_tensorcnt(0);
  }
  __syncthreads();

  // ---------------- GEMM2: out = h @ Wb + bb ----------------
  constexpr int NT2 = COUT / 16;
  v8f acc2[NT2];
#pragma unroll
  for (int nt = 0; nt < NT2; ++nt) acc2[nt] = (v8f)0.f;

  for (int kt = 0; kt < DIM / 4; ++kt) {
    const int k0 = kt * 4 + half * 2;
    v2f a;
    a.x = sH[(wave * 16 + l16) * DIM + k0];
    a.y = sH[(wave * 16 + l16) * DIM + k0 + 1];
#pragma unroll
    for (int nt = 0; nt < NT2; ++nt) {
      v2f b;
      b.x = sW[k0 * COUT + nt * 16 + l16];
      b.y = sW[(k0 + 1) * COUT + nt * 16 + l16];
      acc2[nt] = __builtin_amdgcn_wmma_f32_16x16x4_f32(
          false, a, false, b, (short)0, acc2[nt], false, false);
    }
  }

#pragma unroll
  for (int nt = 0; nt < NT2; ++nt) {
    const int col = nt * 16 + l16;
    const float bias = bb[col];
#pragma unroll
    for (int i = 0; i < 8; ++i) {
      const int row = rowBase + half * 8 + i;
      if (row < n) {
        float v = acc2[nt][i] + bias;
        if (RELU_OUT) v = fmaxf(v, 0.f);
        out[(size_t)row * COUT + col] = v;
      }
    }
  }
}

// ---------------------------------------------------------------------------
extern "C" void kernel_launch(void* const* d_in, const int* in_sizes, int n_in,
                              void* d_out, int out_size, void* d_ws, size_t ws_size,
                              hipStream_t stream) {
  const int*   xi  = (const int*)d_in[0];
  const int*   ei  = (const int*)d_in[1];
  const float* emb = (const float*)d_in[2];
  const float* W1a = (const float*)d_in[3];
  const float* b1a = (const float*)d_in[4];
  const float* W1b = (const float*)d_in[5];
  const float* b1b = (const float*)d_in[6];
  const float* W2a = (const float*)d_in[7];
  const float* b2a = (const float*)d_in[8];
  const float* W2b = (const float*)d_in[9];
  const float* b2b = (const float*)d_in[10];
  float* out = (float*)d_out;

  const int N = in_sizes[0];
  const int E = in_sizes[1] / 2;

  // Workspace: two N x 128 fp32 buffers (51.2 MB total for N=50k).
  float* buf0 = (float*)d_ws;              // x, later x1
  float* buf1 = buf0 + (size_t)N * DIM;    // agg1, later agg2

  const int threads = 256;

  // 1. buf0 = emb[xi]; buf1 = buf0   (agg1 init)
  gather_init<<<(N * 32 + threads - 1) / threads, threads, 0, stream>>>(
      emb, xi, buf0, buf1, N);

  // 2. buf1[dst] += buf0[src]        (agg1)
  scatter_add<<<(E * 32 + threads - 1) / threads, threads, 0, stream>>>(
      buf0, ei, buf1, E);

  // 3. buf0 = relu(MLP1(buf1))       (x1 overwrites x)
  gin_mlp<128, true><<<(N + 127) / 128, threads, 0, stream>>>(
      buf1, W1a, b1a, W1b, b1b, buf0, N);

  // 4. buf1 = buf0                   (agg2 init)
  copy_f4<<<(N * DIM / 4 + threads - 1) / threads, threads, 0, stream>>>(
      (const float4*)buf0, (float4*)buf1, N * DIM / 4);

  // 5. buf1[dst] += buf0[src]        (agg2)
  scatter_add<<<(E * 32 + threads - 1) / threads, threads, 0, stream>>>(
      buf0, ei, buf1, E);

  // 6. out = MLP2(buf1)              [N,64]
  gin_mlp<64, false><<<(N + 127) / 128, threads, 0, stream>>>(
      buf1, W2a, b2a, W2b, b2b, out, N);
}